// BailingMoeV2SparseMoeBlock_23948737642888
// MI455X (gfx1250) — compile-verified
//
#include <hip/hip_runtime.h>
#include <hip/hip_bf16.h>

// Problem constants (match reference)
#define TOK   1024
#define HID   2048
#define IMD   1024
#define NEXP  32
#define NGRP  8
#define EPG   (NEXP / NGRP)   // 4
#define TOPKG 4
#define TOPK  8
#define RSCALE 2.5f
// padded rows: sum ceil(count_e/16)*16 <= 8672; + 64-row M-block overhang -> 8960
#define MAXROWS 8960

#define KC    64              // K-chunk staged in LDS per double-buffer step
#define APAD  68              // LDS row stride in floats (272B = 17*16B: b128-aligned,
                              // and 4-bank row advance -> conflict-free ds_load_b64)

typedef __attribute__((ext_vector_type(2))) float v2f;
typedef __attribute__((ext_vector_type(8))) float v8f;

__device__ __forceinline__ float sigmoidf_(float x) { return 1.0f / (1.0f + __expf(-x)); }
__device__ __forceinline__ float siluf_(float x)    { return x   / (1.0f + __expf(-x)); }

// CDNA5 async global->LDS DMA (ASYNCcnt-tracked). VDST = per-lane LDS byte
// offset VGPR, VADDR = 64-bit global address, saddr = off (NULL).
__device__ __forceinline__ void async_load_b128(unsigned lds_off, const float* gaddr) {
    asm volatile("global_load_async_to_lds_b128 %0, %1, off"
                 :: "v"(lds_off), "v"(gaddr) : "memory");
}
__device__ __forceinline__ void wait_asynccnt0() {
    asm volatile("s_wait_asynccnt 0x0" ::: "memory");
}

// ---------------------------------------------------------------------------
// 1) Router: one block per token (256 threads).
// ---------------------------------------------------------------------------
__global__ void router_kernel(const float* __restrict__ x,
                              const float* __restrict__ gate_w,
                              const float* __restrict__ bias,
                              int*  __restrict__ counts,
                              int*  __restrict__ list,   // [NEXP][TOK]
                              int*  __restrict__ ids,    // [TOK][TOPK]
                              int*  __restrict__ pos,    // [TOK][TOPK]
                              float* __restrict__ wts)   // [TOK][TOPK]
{
    const int t   = blockIdx.x;
    const int tid = threadIdx.x;               // 256 threads
    __shared__ float red[256];
    __shared__ float sc[NEXP];                 // sigmoid scores
    __shared__ float sr[NEXP];                 // scores + bias (routing)

    const int e = tid >> 3, seg = tid & 7;     // 32 experts x 8 segments
    const float* xr = x + (size_t)t * HID + seg * (HID / 8);
    const float* gr = gate_w + (size_t)e * HID + seg * (HID / 8);
    float s = 0.f;
    #pragma unroll 4
    for (int k = 0; k < HID / 8; ++k) s += xr[k] * gr[k];
    red[tid] = s;
    __syncthreads();

    if (tid < NEXP) {
        float l = 0.f;
        #pragma unroll
        for (int j = 0; j < 8; ++j) l += red[tid * 8 + j];
        float sg = sigmoidf_(l);
        sc[tid] = sg;
        sr[tid] = sg + bias[tid];
    }
    __syncthreads();

    if (tid == 0) {
        float gs[NGRP];
        for (int g = 0; g < NGRP; ++g) {       // group score = top-2 sum
            float m1 = -3.4e38f, m2 = -3.4e38f;
            for (int j = 0; j < EPG; ++j) {
                float v = sr[g * EPG + j];
                if (v > m1) { m2 = m1; m1 = v; }
                else if (v > m2) { m2 = v; }
            }
            gs[g] = m1 + m2;
        }
        bool gsel[NGRP];
        for (int g = 0; g < NGRP; ++g) gsel[g] = false;
        for (int it = 0; it < TOPKG; ++it) {   // top-4 groups (ties -> lowest idx)
            float best = -3.4e38f; int bi = 0;
            for (int g = 0; g < NGRP; ++g)
                if (!gsel[g] && gs[g] > best) { best = gs[g]; bi = g; }
            gsel[bi] = true;
        }
        bool used[NEXP];
        for (int i = 0; i < NEXP; ++i) used[i] = false;
        int   sel[TOPK]; float w[TOPK]; float sum = 0.f;
        for (int it = 0; it < TOPK; ++it) {    // top-8 masked experts
            float best = -3.4e38f; int bi = 0;
            for (int i = 0; i < NEXP; ++i)
                if (gsel[i / EPG] && !used[i] && sr[i] > best) { best = sr[i]; bi = i; }
            used[bi] = true;
            sel[it] = bi;
            w[it]   = sc[bi];                  // UNBIASED sigmoid score
            sum    += sc[bi];
        }
        const float inv = RSCALE / (sum + 1e-20f);
        for (int it = 0; it < TOPK; ++it) {
            const int eid = sel[it];
            const int p   = atomicAdd(&counts[eid], 1);   // int atomic only
            list[eid * TOK + p] = t;
            ids[t * TOPK + it] = eid;
            pos[t * TOPK + it] = p;
            wts[t * TOPK + it] = w[it] * inv;
        }
    }
}

// ---------------------------------------------------------------------------
// 2) 16-padded exclusive prefix sum of counts -> row bases
// ---------------------------------------------------------------------------
__global__ void offsets_kernel(const int* __restrict__ counts, int* __restrict__ prow)
{
    if (threadIdx.x == 0) {
        int acc = 0;
        for (int e = 0; e < NEXP; ++e) {
            prow[e] = acc;
            acc += ((counts[e] + 15) >> 4) << 4;
        }
        prow[NEXP] = acc;
    }
}

// ---------------------------------------------------------------------------
// 3) gate_up GEMM + SiLU*up. WMMA f32 16x16x4, M blocked by 64.
//    A tile (64 gathered token rows) is shared by all 8 waves -> staged into
//    LDS with async global->LDS DMA, double-buffered: DMA of chunk c+1
//    overlaps WMMA on chunk c; s_wait_asynccnt 0 precedes the barrier so all
//    waves' DMA has landed before anyone consumes/overwrites a buffer.
// ---------------------------------------------------------------------------
__global__ __launch_bounds__(256)
void gateup_kernel(const float* __restrict__ x,
                   const float* __restrict__ wgu,    // [slot][2*IMD][HID] row-major
                   const int*   __restrict__ list,
                   const int*   __restrict__ counts, int countConst,
                   const int*   __restrict__ prow,
                   float* __restrict__ hout)         // rows x IMD
{
    __shared__ float atile[2][64 * APAD];            // ~34 KB

    const int e     = blockIdx.z;
    const int count = counts ? counts[e] : countConst;
    const int mbase = blockIdx.x * 64;
    if (mbase >= count) return;                      // block-uniform

    const int tid  = threadIdx.x;
    const int wave = tid >> 5;
    const int lane = tid & 31;
    const int mrow = lane & 15;                      // A row (M) / B row (N) / C col
    const int kh   = lane >> 4;                      // K-pair selector
    const int koff = 2 * kh;
    const int n0   = blockIdx.y * 128 + wave * 16;

    // B: W^T -> per-lane row (n0+mrow) of the row-major weight
    const float* Wg = wgu + (size_t)e * (2 * IMD) * HID + (size_t)(n0 + mrow) * HID + koff;
    const float* Wu = Wg + (size_t)IMD * HID;

    // Staging role: thread copies row (tid&63), 4 consecutive b128 segments.
    const int srow = tid & 63;
    const int sseg = (tid >> 6) * 4;                 // of 16 segments per row
    {
        // nothing: keep scope tight below
    }
    const int ridx = mbase + srow;
    const int cidx = ridx < count ? ridx : count - 1;     // clamp padded rows
    const int stok = list ? list[e * TOK + cidx] : cidx;
    const float* sgp = x + (size_t)stok * HID + sseg * 4; // + kc per chunk
    unsigned slds[2];
    slds[0] = (unsigned)(unsigned long long)&atile[0][srow * APAD + sseg * 4];
    slds[1] = (unsigned)(unsigned long long)&atile[1][srow * APAD + sseg * 4];

    auto stage = [&](int kc, int buf) {
        const float* g = sgp + kc;
        const unsigned l = slds[buf];
        #pragma unroll
        for (int i = 0; i < 4; ++i)
            async_load_b128(l + i * 16, g + i * 4);
    };

    // prime buffer 0
    stage(0, 0);
    wait_asynccnt0();
    __syncthreads();

    v8f accG[4] = {}; v8f accU[4] = {};
    const int nch = HID / KC;                        // 32 chunks
    for (int c = 0; c < nch; ++c) {
        const int kc = c * KC;
        if (c + 1 < nch) {
            stage(kc + KC, (c + 1) & 1);             // async DMA next chunk
            __builtin_prefetch(Wg + kc + KC, 0, 3);  // -> global_prefetch_b8
            __builtin_prefetch(Wu + kc + KC, 0, 3);
        }
        const float* at = &atile[c & 1][0];
        #pragma unroll 4
        for (int k0 = 0; k0 < KC; k0 += 4) {
            v2f bg = *(const v2f*)(Wg + kc + k0);
            v2f bu = *(const v2f*)(Wu + kc + k0);
            #pragma unroll
            for (int mt = 0; mt < 4; ++mt) {
                v2f a = *(const v2f*)(at + (mt * 16 + mrow) * APAD + k0 + koff); // ds_load_b64
                accG[mt] = __builtin_amdgcn_wmma_f32_16x16x4_f32(false, a, false, bg,
                                                                 (short)0, accG[mt], false, false);
                accU[mt] = __builtin_amdgcn_wmma_f32_16x16x4_f32(false, a, false, bu,
                                                                 (short)0, accU[mt], false, false);
            }
        }
        wait_asynccnt0();                            // own DMA done before barrier
        __syncthreads();                             // all waves' DMA landed / reads done
    }

    const int hrow0 = (prow ? prow[e] : 0) + mbase;
    const int col   = n0 + mrow;
    #pragma unroll
    for (int mt = 0; mt < 4; ++mt) {
        if (mbase + mt * 16 < count) {               // uniform store guard
            #pragma unroll
            for (int r = 0; r < 8; ++r) {
                const int m = mt * 16 + r + 8 * kh;  // C layout: reg r, lanes16-31 -> M=r+8
                hout[(size_t)(hrow0 + m) * IMD + col] = siluf_(accG[mt][r]) * accU[mt][r];
            }
        }
    }
}

// ---------------------------------------------------------------------------
// 4) down-proj GEMM, M blocked by 64, same async-LDS A staging
//    (A rows contiguous). Routed -> eout rows; shared -> y rows.
// ---------------------------------------------------------------------------
__global__ __launch_bounds__(256)
void down_kernel(const float* __restrict__ hin,      // rows x IMD
                 const float* __restrict__ wd,       // [slot][HID][IMD] row-major
                 const int*   __restrict__ counts, int countConst,
                 const int*   __restrict__ prow,
                 float* __restrict__ out)            // rows x HID
{
    __shared__ float atile[2][64 * APAD];

    const int e     = blockIdx.z;
    const int count = counts ? counts[e] : countConst;
    const int mbase = blockIdx.x * 64;
    if (mbase >= count) return;

    const int tid  = threadIdx.x;
    const int wave = tid >> 5;
    const int lane = tid & 31;
    const int mrow = lane & 15;
    const int kh   = lane >> 4;
    const int koff = 2 * kh;
    const int n0   = blockIdx.y * 128 + wave * 16;

    const int base = (prow ? prow[e] : 0) + mbase;
    const float* Wr = wd + (size_t)e * HID * IMD + (size_t)(n0 + mrow) * IMD + koff;

    const int srow = tid & 63;
    const int sseg = (tid >> 6) * 4;
    const float* sgp = hin + (size_t)(base + srow) * IMD + sseg * 4; // overhang rows exist in ws
    unsigned slds[2];
    slds[0] = (unsigned)(unsigned long long)&atile[0][srow * APAD + sseg * 4];
    slds[1] = (unsigned)(unsigned long long)&atile[1][srow * APAD + sseg * 4];

    auto stage = [&](int kc, int buf) {
        const float* g = sgp + kc;
        const unsigned l = slds[buf];
        #pragma unroll
        for (int i = 0; i < 4; ++i)
            async_load_b128(l + i * 16, g + i * 4);
    };

    stage(0, 0);
    wait_asynccnt0();
    __syncthreads();

    v8f acc[4] = {};
    const int nch = IMD / KC;                        // 16 chunks
    for (int c = 0; c < nch; ++c) {
        const int kc = c * KC;
        if (c + 1 < nch) {
            stage(kc + KC, (c + 1) & 1);
            __builtin_prefetch(Wr + kc + KC, 0, 3);  // -> global_prefetch_b8
        }
        const float* at = &atile[c & 1][0];
        #pragma unroll 4
        for (int k0 = 0; k0 < KC; k0 += 4) {
            v2f b = *(const v2f*)(Wr + kc + k0);
            #pragma unroll
            for (int mt = 0; mt < 4; ++mt) {
                v2f a = *(const v2f*)(at + (mt * 16 + mrow) * APAD + k0 + koff); // ds_load_b64
                acc[mt] = __builtin_amdgcn_wmma_f32_16x16x4_f32(false, a, false, b,
                                                                (short)0, acc[mt], false, false);
            }
        }
        wait_asynccnt0();
        __syncthreads();
    }

    const int col = n0 + mrow;
    #pragma unroll
    for (int mt = 0; mt < 4; ++mt) {
        if (mbase + mt * 16 < count) {
            #pragma unroll
            for (int r = 0; r < 8; ++r) {
                const int m = mt * 16 + r + 8 * kh;
                out[(size_t)(base + m) * HID + col] = acc[mt][r];
            }
        }
    }
}

// ---------------------------------------------------------------------------
// 5) Combine: y[t] += sum_k w_k * eout[row(e_k, t)]  (fixed k order -> deterministic)
// ---------------------------------------------------------------------------
__global__ void combine_kernel(const float* __restrict__ eout,
                               const int*   __restrict__ ids,
                               const int*   __restrict__ pos,
                               const float* __restrict__ wts,
                               const int*   __restrict__ prow,
                               float* __restrict__ y)
{
    const int t = blockIdx.x;
    __shared__ int   srow[TOPK];
    __shared__ float sw[TOPK];
    if (threadIdx.x < TOPK) {
        const int k = threadIdx.x;
        const int e = ids[t * TOPK + k];
        srow[k] = prow[e] + pos[t * TOPK + k];
        sw[k]   = wts[t * TOPK + k];
    }
    __syncthreads();
    for (int c = threadIdx.x; c < HID; c += blockDim.x) {
        float acc = y[(size_t)t * HID + c];
        #pragma unroll
        for (int k = 0; k < TOPK; ++k)
            acc += sw[k] * eout[(size_t)srow[k] * HID + c];
        y[(size_t)t * HID + c] = acc;
    }
}

// ---------------------------------------------------------------------------
// Launch
// ---------------------------------------------------------------------------
extern "C" void kernel_launch(void* const* d_in, const int* in_sizes, int n_in,
                              void* d_out, int out_size, void* d_ws, size_t ws_size,
                              hipStream_t stream)
{
    const float* x   = (const float*)d_in[0];   // [TOK,HID]
    const float* gw  = (const float*)d_in[1];   // [NEXP,HID]
    const float* eb  = (const float*)d_in[2];   // [NEXP]
    const float* wgu = (const float*)d_in[3];   // [NEXP,2*IMD,HID]
    const float* wd  = (const float*)d_in[4];   // [NEXP,HID,IMD]
    const float* sgu = (const float*)d_in[5];   // [2*IMD,HID]
    const float* sd  = (const float*)d_in[6];   // [HID,IMD]
    float* y = (float*)d_out;

    char* ws = (char*)d_ws;
    size_t off = 0;
    auto alloc = [&](size_t bytes) -> void* {
        void* p = ws + off;
        off = (off + bytes + 255) & ~(size_t)255;
        return p;
    };
    int*   counts = (int*)  alloc(NEXP * 4);
    int*   prow   = (int*)  alloc((NEXP + 1) * 4);
    int*   list   = (int*)  alloc((size_t)NEXP * TOK * 4);
    int*   ids    = (int*)  alloc((size_t)TOK * TOPK * 4);
    int*   pos    = (int*)  alloc((size_t)TOK * TOPK * 4);
    float* wts    = (float*)alloc((size_t)TOK * TOPK * 4);
    float* h_all  = (float*)alloc((size_t)MAXROWS * IMD * 4);   // ~37 MB
    float* eout   = (float*)alloc((size_t)MAXROWS * HID * 4);   // ~73 MB
    float* hs     = (float*)alloc((size_t)TOK * IMD * 4);       //  4 MB

    hipMemsetAsync(counts, 0, NEXP * 4, stream);
    router_kernel<<<TOK, 256, 0, stream>>>(x, gw, eb, counts, list, ids, pos, wts);
    offsets_kernel<<<1, 32, 0, stream>>>(counts, prow);

    // routed experts: gate_up + SiLU*up -> h_all
    gateup_kernel<<<dim3(TOK / 64, IMD / 128, NEXP), 256, 0, stream>>>(
        x, wgu, list, counts, 0, prow, h_all);
    // shared expert: gate_up + SiLU*up -> hs
    gateup_kernel<<<dim3(TOK / 64, IMD / 128, 1), 256, 0, stream>>>(
        x, sgu, nullptr, nullptr, TOK, nullptr, hs);
    // shared expert down-proj writes every y element (initializes y)
    down_kernel<<<dim3(TOK / 64, HID / 128, 1), 256, 0, stream>>>(
        hs, sd, nullptr, TOK, nullptr, y);
    // routed down-proj -> eout rows
    down_kernel<<<dim3(TOK / 64, HID / 128, NEXP), 256, 0, stream>>>(
        h_all, wd, counts, 0, prow, eout);
    // weighted combine into y
    combine_kernel<<<TOK, 256, 0, stream>>>(eout, ids, pos, wts, prow, y);

    (void)in_sizes; (void)n_in; (void)out_size; (void)ws_size;
}